// GNN_74105365725675
// MI455X (gfx1250) — compile-verified
//
#include <hip/hip_runtime.h>
#include <math.h>

#define N_NODES 100000
#define N_EDGES 3200000
#define IN_DIM  64
#define H1      16

typedef __attribute__((ext_vector_type(2))) float v2f;
typedef __attribute__((ext_vector_type(8))) float v8f;

// ---------------- init / degree ----------------

__global__ void fill_kernel(float* __restrict__ p, float v, int n) {
    for (int i = blockIdx.x * blockDim.x + threadIdx.x; i < n; i += gridDim.x * blockDim.x)
        p[i] = v;
}

__global__ void deg_kernel(const int* __restrict__ dst, const float* __restrict__ w,
                           float* __restrict__ deg) {
    int e = blockIdx.x * blockDim.x + threadIdx.x;
    if (e < N_EDGES) atomicAdd(&deg[dst[e]], w[e]);
}

__global__ void dinv_kernel(float* __restrict__ deg) {
    int i = blockIdx.x * blockDim.x + threadIdx.x;
    if (i < N_NODES) {
        float d = deg[i];
        deg[i] = (d > 0.0f) ? rsqrtf(d) : 0.0f;   // in place: deg -> dinv
    }
}

// ---------------- layer 1: h = x @ W1 via fp32 WMMA 16x16x4 ----------------
// One wave computes one 16x16 output tile; K=64 done as 16 chained k=4 WMMAs.
// A (16x4 f32): lane m=L%16, VGPR v holds K=(L/16)*2+v
// B (4x16 f32): lane n=L%16, VGPR v holds K=(L/16)*2+v
// D (16x16 f32): VGPR v: lanes 0-15 -> M=v, lanes 16-31 -> M=v+8

__global__ void gemm1_wmma(const float* __restrict__ x, const float* __restrict__ W1,
                           float* __restrict__ h) {
    int wave = (blockIdx.x * blockDim.x + threadIdx.x) >> 5;   // tile id (wave-uniform)
    int lane = threadIdx.x & 31;
    if (wave >= N_NODES / 16) return;    // whole-wave exit: EXEC stays all-1s for WMMA

    int mn = lane & 15;                  // row of A / col of B,D
    int kh = lane >> 4;                  // which K-half this lane holds

    const float* xrow = x + (wave * 16 + mn) * IN_DIM;
    v8f c = {};
    #pragma unroll
    for (int kk = 0; kk < IN_DIM / 4; ++kk) {
        int kb = kk * 4 + kh * 2;
        v2f a; a.x = xrow[kb];             a.y = xrow[kb + 1];
        v2f b; b.x = W1[kb * H1 + mn];     b.y = W1[(kb + 1) * H1 + mn];
        c = __builtin_amdgcn_wmma_f32_16x16x4_f32(
                /*neg_a=*/false, a, /*neg_b=*/false, b,
                /*c_mod=*/(short)0, c, /*reuse_a=*/false, /*reuse_b=*/false);
    }
    #pragma unroll
    for (int v = 0; v < 8; ++v)
        h[(wave * 16 + v + kh * 8) * H1 + mn] = c[v];
}

// ---------------- edge scatter, layer 1: 16 lanes per edge ----------------

__global__ void agg1_edges(const int* __restrict__ src, const int* __restrict__ dst,
                           const float* __restrict__ w, const float* __restrict__ dinv,
                           const float* __restrict__ h, float* __restrict__ agg) {
    long long t = (long long)blockIdx.x * blockDim.x + threadIdx.x;
    long long e = t >> 4;
    int j = (int)(t & 15);
    if (e >= N_EDGES) return;
    int s = src[e], d = dst[e];
    float coef = dinv[s] * w[e] * dinv[d];
    atomicAdd(&agg[(long long)d * H1 + j], h[(long long)s * H1 + j] * coef);
}

// self-loop + bias + ReLU, in place on agg1
__global__ void post1_kernel(const float* __restrict__ h, const float* __restrict__ dinv,
                             const float* __restrict__ b1, float* __restrict__ agg1) {
    int idx = blockIdx.x * blockDim.x + threadIdx.x;
    if (idx >= N_NODES * H1) return;
    int i = idx >> 4, j = idx & 15;
    float di = dinv[i];
    float v = agg1[idx] + h[idx] * di * di + b1[j];
    agg1[idx] = v > 0.0f ? v : 0.0f;
}

// ---------------- layer 2: h2 = h1 @ W2 (16 -> 2), scalar ----------------

__global__ void gemm2_kernel(const float* __restrict__ h1, const float* __restrict__ W2,
                             float* __restrict__ h2) {
    int i = blockIdx.x * blockDim.x + threadIdx.x;
    if (i >= N_NODES) return;
    float a0 = 0.0f, a1 = 0.0f;
    #pragma unroll
    for (int k = 0; k < H1; ++k) {
        float hv = h1[i * H1 + k];
        a0 = fmaf(hv, W2[k * 2 + 0], a0);
        a1 = fmaf(hv, W2[k * 2 + 1], a1);
    }
    h2[i * 2 + 0] = a0;
    h2[i * 2 + 1] = a1;
}

__global__ void agg2_edges(const int* __restrict__ src, const int* __restrict__ dst,
                           const float* __restrict__ w, const float* __restrict__ dinv,
                           const float* __restrict__ h2, float* __restrict__ agg2) {
    int e = blockIdx.x * blockDim.x + threadIdx.x;
    if (e >= N_EDGES) return;
    int s = src[e], d = dst[e];
    float coef = dinv[s] * w[e] * dinv[d];
    atomicAdd(&agg2[d * 2 + 0], h2[s * 2 + 0] * coef);
    atomicAdd(&agg2[d * 2 + 1], h2[s * 2 + 1] * coef);
}

// self-loop + b2 -> x2; ReLU -> linear -> sigmoid -> out
__global__ void final_kernel(const float* __restrict__ h2, const float* __restrict__ agg2,
                             const float* __restrict__ dinv, const float* __restrict__ b2,
                             const float* __restrict__ lin_w, const float* __restrict__ lin_b,
                             float* __restrict__ out) {
    int i = blockIdx.x * blockDim.x + threadIdx.x;
    if (i >= N_NODES) return;
    float di = dinv[i], d2 = di * di;
    float x20 = agg2[i * 2 + 0] + h2[i * 2 + 0] * d2 + b2[0];
    float x21 = agg2[i * 2 + 1] + h2[i * 2 + 1] * d2 + b2[1];
    float r0 = fmaxf(x20, 0.0f), r1 = fmaxf(x21, 0.0f);
    float z = fmaf(r0, lin_w[0], fmaf(r1, lin_w[1], lin_b[0]));
    out[i] = 1.0f / (1.0f + __expf(-z));          // output 0: sigmoid head [N,1]
    out[N_NODES + i * 2 + 0] = x20;               // output 1: x2 [N,2]
    out[N_NODES + i * 2 + 1] = x21;
}

// ---------------- launch ----------------

extern "C" void kernel_launch(void* const* d_in, const int* in_sizes, int n_in,
                              void* d_out, int out_size, void* d_ws, size_t ws_size,
                              hipStream_t stream) {
    const float* x   = (const float*)d_in[0];
    const int*   ei  = (const int*)  d_in[1];   // [2, E] int32 (JAX x64 disabled)
    const float* w   = (const float*)d_in[2];
    const float* W1  = (const float*)d_in[3];
    const float* b1  = (const float*)d_in[4];
    const float* W2  = (const float*)d_in[5];
    const float* b2  = (const float*)d_in[6];
    const float* lw  = (const float*)d_in[7];
    const float* lb  = (const float*)d_in[8];
    const int* src = ei;
    const int* dst = ei + N_EDGES;

    float* ws   = (float*)d_ws;
    float* deg  = ws;                            // N floats (becomes dinv in place)
    float* h    = deg  + N_NODES;                // 16N floats
    float* agg1 = h    + N_NODES * H1;           // 16N floats (becomes h1 in place)
    float* h2   = agg1 + N_NODES * H1;           // 2N floats
    float* agg2 = h2   + N_NODES * 2;            // 2N floats
    float* out  = (float*)d_out;

    const int B = 256;

    // re-init every call (atomics accumulate; harness does not re-poison)
    fill_kernel<<<1024, B, 0, stream>>>(deg,  1.0f, N_NODES);       // self-loop weight
    fill_kernel<<<2048, B, 0, stream>>>(agg1, 0.0f, N_NODES * H1);
    fill_kernel<<<1024, B, 0, stream>>>(agg2, 0.0f, N_NODES * 2);

    deg_kernel <<<(N_EDGES + B - 1) / B, B, 0, stream>>>(dst, w, deg);
    dinv_kernel<<<(N_NODES + B - 1) / B, B, 0, stream>>>(deg);

    // layer 1
    {
        int waves = N_NODES / 16;                      // 6250 tiles, 1 wave each
        int threads = waves * 32;
        gemm1_wmma<<<(threads + B - 1) / B, B, 0, stream>>>(x, W1, h);
    }
    {
        long long t = (long long)N_EDGES * 16;
        agg1_edges<<<(unsigned)((t + B - 1) / B), B, 0, stream>>>(src, dst, w, deg, h, agg1);
    }
    post1_kernel<<<(N_NODES * H1 + B - 1) / B, B, 0, stream>>>(h, deg, b1, agg1);

    // layer 2 + head
    gemm2_kernel<<<(N_NODES + B - 1) / B, B, 0, stream>>>(agg1, W2, h2);
    agg2_edges  <<<(N_EDGES + B - 1) / B, B, 0, stream>>>(src, dst, w, deg, h2, agg2);
    final_kernel<<<(N_NODES + B - 1) / B, B, 0, stream>>>(h2, agg2, deg, b2, lw, lb, out);
}